// GNNLoss_63840393887854
// MI455X (gfx1250) — compile-verified
//
#include <hip/hip_runtime.h>
#include <hip/hip_bf16.h>
#include <math.h>

// ---------------------------------------------------------------------------
// GNNLoss pipeline for MI455X (gfx1250, wave32).
//  - fp32 WMMA (v_wmma_f32_16x16x4_f32) for the four dense GEMMs
//  - memory-bound gather+online-logsumexp kernel for the 256x4096 "neg" pass
// ---------------------------------------------------------------------------

typedef float v2f __attribute__((ext_vector_type(2)));
typedef float v8f __attribute__((ext_vector_type(8)));

#define B_ROWS   256
#define C_DIM    128
#define NCLS     100
#define KNN_K    8
#define KNEG     4096
#define INV_T    (1.0f / 0.07f)
#define NEG_INF  (-3.402823466e+38f)

// ---------------------------------------------------------------------------
// d_out zeroing (atomicAdd accumulator)
// ---------------------------------------------------------------------------
__global__ void zero_kernel(float* out) {
    if (threadIdx.x == 0 && blockIdx.x == 0) out[0] = 0.0f;
}

// ---------------------------------------------------------------------------
// GEMM + bias + row L2-norm.
//   out[m][n] = l2norm_row( sum_k X[m][k] * B[k][n] + bias[n] )
//   BT=true : B[k][n] = W[n*K + k]   (W stored [128, K], i.e. X @ W^T)
//   BT=false: B[k][n] = W[k*128 + n] (W stored [K, 128], i.e. X @ W)
// Grid: 16 blocks (one per 16-row band), 256 threads (8 waves, one 16x16
// N-tile per wave). Uses V_WMMA_F32_16X16X4_F32 per the CDNA5 ISA layouts:
//   A (16x4 f32): lanes 0-15 -> M=lane, VGPR0/1 = K=0/1; lanes 16-31 -> K=2/3
//   B (4x16 f32): symmetric (lane = N, half selects K pair)
//   C/D (16x16 f32): VGPR r: lanes 0-15 -> M=r, lanes 16-31 -> M=r+8
// ---------------------------------------------------------------------------
template <bool BT>
__global__ __launch_bounds__(256) void gemm_l2norm_kernel(
    const float* __restrict__ X, const float* __restrict__ W,
    const float* __restrict__ bias, float* __restrict__ out, int K) {
    __shared__ float tile[16][132];   // pad: stride 132 % 64 banks = 4
    __shared__ float rnorm[16];

    const int tid  = threadIdx.x;
    const int wave = tid >> 5;        // 0..7  -> N tile
    const int lane = tid & 31;
    const int half = lane >> 4;       // K-pair selector
    const int l16  = lane & 15;
    const int band = blockIdx.x;      // 16-row band
    const int m    = band * 16 + l16; // A row for this lane
    const int n    = wave * 16 + l16; // B column for this lane

    v8f acc = {0.f, 0.f, 0.f, 0.f, 0.f, 0.f, 0.f, 0.f};

    const float* xrow = X + (size_t)m * K;
    #pragma unroll 4
    for (int k0 = 0; k0 < K; k0 += 4) {
        const int ka = k0 + 2 * half;
        v2f a = *(const v2f*)(xrow + ka);
        v2f b;
        if (BT) {
            b = *(const v2f*)(W + (size_t)n * K + ka);
        } else {
            b.x = W[(size_t)ka * C_DIM + n];
            b.y = W[(size_t)(ka + 1) * C_DIM + n];
        }
        acc = __builtin_amdgcn_wmma_f32_16x16x4_f32(
            /*neg_a=*/false, a, /*neg_b=*/false, b,
            /*c_mod=*/(short)0, acc, /*reuse_a=*/false, /*reuse_b=*/false);
    }

    const float bv = bias[n];
    #pragma unroll
    for (int r = 0; r < 8; ++r) {
        const int rl = r + 8 * half;  // local row per C/D layout
        tile[rl][n] = acc[r] + bv;
    }
    __syncthreads();

    if (tid < 16) {
        float s = 0.f;
        for (int c = 0; c < C_DIM; ++c) { float v = tile[tid][c]; s += v * v; }
        rnorm[tid] = rsqrtf(s);
    }
    __syncthreads();

    for (int i = tid; i < 16 * C_DIM; i += 256) {
        const int r = i >> 7, c = i & 127;
        out[(size_t)(band * 16 + r) * C_DIM + c] = tile[r][c] * rnorm[r];
    }
}

// ---------------------------------------------------------------------------
// Row softmax of logits [256, NCLS] + L2 norm of the softmax row.
// Grid 256, block 128.
// ---------------------------------------------------------------------------
__global__ __launch_bounds__(128) void softmax_norm_kernel(
    const float* __restrict__ logits, float* __restrict__ soft,
    float* __restrict__ wnorm) {
    __shared__ float red[128];
    const int row = blockIdx.x, t = threadIdx.x;

    float v = (t < NCLS) ? logits[(size_t)row * NCLS + t] : NEG_INF;
    red[t] = v; __syncthreads();
    for (int s = 64; s > 0; s >>= 1) {
        if (t < s) red[t] = fmaxf(red[t], red[t + s]);
        __syncthreads();
    }
    const float mx = red[0]; __syncthreads();

    const float e = (t < NCLS) ? __expf(v - mx) : 0.f;
    red[t] = e; __syncthreads();
    for (int s = 64; s > 0; s >>= 1) {
        if (t < s) red[t] += red[t + s];
        __syncthreads();
    }
    const float inv = 1.0f / red[0]; __syncthreads();

    const float p = e * inv;
    if (t < NCLS) soft[(size_t)row * NCLS + t] = p;

    red[t] = p * p; __syncthreads();
    for (int s = 64; s > 0; s >>= 1) {
        if (t < s) red[t] += red[t + s];
        __syncthreads();
    }
    if (t == 0) wnorm[row] = sqrtf(red[0]);
}

// ---------------------------------------------------------------------------
// kNN on softmax rows: top-8 of -dist  (self always picked via diag trick).
// Grid 256 (row i), block 256 (candidate j).
// ---------------------------------------------------------------------------
__global__ __launch_bounds__(256) void knn_kernel(
    const float* __restrict__ soft, const float* __restrict__ wnorm,
    int* __restrict__ nbr) {
    __shared__ float srow[NCLS];
    __shared__ float base[256];
    __shared__ float rv[256];
    __shared__ int   ri[256];

    const int i = blockIdx.x, j = threadIdx.x;
    for (int c = j; c < NCLS; c += 256) srow[c] = soft[(size_t)i * NCLS + c];
    __syncthreads();

    float d = 0.f;
    const float* srj = soft + (size_t)j * NCLS;
    for (int c = 0; c < NCLS; ++c) d += srow[c] * srj[c];
    const float denom = fmaxf(wnorm[i] * wnorm[j], 1e-7f);
    // -dist: diag -> +2 (always max); off-diag -> sim - 1
    base[j] = (j == i) ? 2.0f : (d / denom - 1.0f);
    __syncthreads();

    for (int t = 0; t < KNN_K; ++t) {
        rv[j] = base[j]; ri[j] = j; __syncthreads();
        for (int s = 128; s > 0; s >>= 1) {
            if (j < s) {
                const float a = rv[j], b = rv[j + s];
                if (b > a || (b == a && ri[j + s] < ri[j])) {
                    rv[j] = b; ri[j] = ri[j + s];
                }
            }
            __syncthreads();
        }
        if (j == 0) { nbr[i * KNN_K + t] = ri[0]; base[ri[0]] = NEG_INF; }
        __syncthreads();
    }
}

// ---------------------------------------------------------------------------
// cat[m] = [ f[m], mean_{t<8} f[nbr[m][t]] ]   -> [256, 256]
// Grid 256, block 128.
// ---------------------------------------------------------------------------
__global__ __launch_bounds__(128) void gnn_cat_kernel(
    const float* __restrict__ f, const int* __restrict__ nbr,
    float* __restrict__ cat) {
    const int m = blockIdx.x, c = threadIdx.x;
    float a = 0.f;
    #pragma unroll
    for (int t = 0; t < KNN_K; ++t)
        a += f[(size_t)nbr[m * KNN_K + t] * C_DIM + c];
    cat[(size_t)m * 256 + c]         = f[(size_t)m * C_DIM + c];
    cat[(size_t)m * 256 + C_DIM + c] = a * 0.125f;
}

// ---------------------------------------------------------------------------
// Smoothed positives:
//   f_sgs = l2norm(0.75*mem_l[idx] + 0.25*f_gs), f_sgt likewise w/ mem_ab,f_gt
//   pos[0]=<mem_ab[idx],f_es> pos[1]=<mem_l[idx],f_et>
//   pos[2]=<f_sgt,f_gs>       pos[3]=<f_sgs,f_gt>
// Grid 256, block 128.
// ---------------------------------------------------------------------------
__global__ __launch_bounds__(128) void smooth_pos_kernel(
    const float* __restrict__ mem_l, const float* __restrict__ mem_ab,
    const int* __restrict__ idx,
    const float* __restrict__ f_es, const float* __restrict__ f_et,
    const float* __restrict__ f_gs, const float* __restrict__ f_gt,
    float* __restrict__ pos) {
    __shared__ float red[128];
    const int m = blockIdx.x, c = threadIdx.x;

    auto bsum = [&](float x) -> float {
        __syncthreads();
        red[c] = x; __syncthreads();
        for (int s = 64; s > 0; s >>= 1) {
            if (c < s) red[c] += red[c + s];
            __syncthreads();
        }
        return red[0];
    };

    const size_t base = (size_t)idx[m] * C_DIM;
    const float ml  = mem_l[base + c];
    const float mab = mem_ab[base + c];
    const float es  = f_es[(size_t)m * C_DIM + c];
    const float et  = f_et[(size_t)m * C_DIM + c];
    const float gs  = f_gs[(size_t)m * C_DIM + c];
    const float gt  = f_gt[(size_t)m * C_DIM + c];

    const float s1 = 0.75f * ml + 0.25f * gs;   // pre-norm f_sgs
    const float s2 = 0.75f * mab + 0.25f * gt;  // pre-norm f_sgt

    const float n1 = rsqrtf(bsum(s1 * s1));
    const float n2 = rsqrtf(bsum(s2 * s2));
    const float ls_pos = bsum(mab * es);
    const float lt_pos = bsum(ml * et);
    const float gs_pos = bsum((s2 * n2) * gs);
    const float gt_pos = bsum((s1 * n1) * gt);

    if (c == 0) {
        pos[0 * B_ROWS + m] = ls_pos;
        pos[1 * B_ROWS + m] = lt_pos;
        pos[2 * B_ROWS + m] = gs_pos;
        pos[3 * B_ROWS + m] = gt_pos;
    }
}

// ---------------------------------------------------------------------------
// Neg pass + loss. Grid 256 (b), block 256 (8 waves).
// Each wave processes one gathered row at a time: coalesced float4 loads of
// mem_l[j] and mem_ab[j], 4 dot products against {f_es,f_gs,f_et,f_gt}[b],
// butterfly reduce, online logsumexp (redundant per lane). Cross-wave combine
// in LDS; positive column folded in; atomicAdd of per-row loss / 256.
// ---------------------------------------------------------------------------
__global__ __launch_bounds__(256) void neg_loss_kernel(
    const float* __restrict__ mem_l, const float* __restrict__ mem_ab,
    const int* __restrict__ cidx,
    const float* __restrict__ f_es, const float* __restrict__ f_et,
    const float* __restrict__ f_gs, const float* __restrict__ f_gt,
    const float* __restrict__ pos, float* __restrict__ out) {
    __shared__ float sm[4][8], ss[4][8];

    const int b    = blockIdx.x;
    const int tid  = threadIdx.x;
    const int wave = tid >> 5;
    const int lane = tid & 31;
    const int c0   = lane * 4;

    const float4 es = *(const float4*)(f_es + (size_t)b * C_DIM + c0);
    const float4 et = *(const float4*)(f_et + (size_t)b * C_DIM + c0);
    const float4 gs = *(const float4*)(f_gs + (size_t)b * C_DIM + c0);
    const float4 gt = *(const float4*)(f_gt + (size_t)b * C_DIM + c0);

    float mS = NEG_INF, sS = 0.f, mT = NEG_INF, sT = 0.f;
    float mGS = NEG_INF, sGS = 0.f, mGT = NEG_INF, sGT = 0.f;

    auto upd = [](float& m, float& s, float v) {
        if (v > m) { s = s * __expf(m - v) + 1.0f; m = v; }
        else       { s += __expf(v - m); }
    };

    const int* crow = cidx + (size_t)b * KNEG;
    for (int k = wave; k < KNEG; k += 8) {
        const size_t j = (size_t)crow[k] * C_DIM;
        const float4 wt = *(const float4*)(mem_l + j + c0);
        const float4 wa = *(const float4*)(mem_ab + j + c0);

        if (k + 8 < KNEG) {  // prefetch next gathered rows
            const size_t jn = (size_t)crow[k + 8] * C_DIM;
            __builtin_prefetch(mem_l + jn + c0, 0, 3);
            __builtin_prefetch(mem_ab + jn + c0, 0, 3);
        }

        float dS  = wt.x * es.x + wt.y * es.y + wt.z * es.z + wt.w * es.w;
        float dGS = wt.x * gs.x + wt.y * gs.y + wt.z * gs.z + wt.w * gs.w;
        float dT  = wa.x * et.x + wa.y * et.y + wa.z * et.z + wa.w * et.w;
        float dGT = wa.x * gt.x + wa.y * gt.y + wa.z * gt.z + wa.w * gt.w;

        #pragma unroll
        for (int off = 16; off > 0; off >>= 1) {
            dS  += __shfl_xor(dS, off);
            dGS += __shfl_xor(dGS, off);
            dT  += __shfl_xor(dT, off);
            dGT += __shfl_xor(dGT, off);
        }

        upd(mS, sS, dS * INV_T);
        upd(mT, sT, dT * INV_T);
        upd(mGS, sGS, dGS * INV_T);
        upd(mGT, sGT, dGT * INV_T);
    }

    if (lane == 0) {
        sm[0][wave] = mS;  ss[0][wave] = sS;
        sm[1][wave] = mT;  ss[1][wave] = sT;
        sm[2][wave] = mGS; ss[2][wave] = sGS;
        sm[3][wave] = mGT; ss[3][wave] = sGT;
    }
    __syncthreads();

    if (tid == 0) {
        float loss = 0.f;
        #pragma unroll
        for (int st = 0; st < 4; ++st) {
            const float p = pos[st * B_ROWS + b] * INV_T;
            float M = p, S = 1.0f;  // positive column at index 0
            for (int w = 0; w < 8; ++w) {
                const float m2 = sm[st][w], s2 = ss[st][w];
                if (m2 > M) { S = S * __expf(M - m2) + s2; M = m2; }
                else        { S += s2 * __expf(m2 - M); }
            }
            loss += M + __logf(S) - p;
        }
        atomicAdd(out, loss * (1.0f / (float)B_ROWS));
    }
}

// ---------------------------------------------------------------------------
// Host launcher
// ---------------------------------------------------------------------------
extern "C" void kernel_launch(void* const* d_in, const int* in_sizes, int n_in,
                              void* d_out, int out_size, void* d_ws,
                              size_t ws_size, hipStream_t stream) {
    (void)in_sizes; (void)n_in; (void)out_size; (void)ws_size;

    const float* f_s    = (const float*)d_in[1];
    const float* l_s    = (const float*)d_in[2];
    const float* f_t    = (const float*)d_in[3];
    const float* l_t    = (const float*)d_in[4];
    const int*   idx    = (const int*)d_in[5];
    const int*   cidx   = (const int*)d_in[6];
    const float* W_es   = (const float*)d_in[7];
    const float* b_es   = (const float*)d_in[8];
    const float* W_et   = (const float*)d_in[9];
    const float* b_et   = (const float*)d_in[10];
    const float* W_gs   = (const float*)d_in[11];
    const float* b_gs   = (const float*)d_in[12];
    const float* W_gt   = (const float*)d_in[13];
    const float* b_gt   = (const float*)d_in[14];
    const float* mem_l  = (const float*)d_in[15];
    const float* mem_ab = (const float*)d_in[16];
    float* out = (float*)d_out;

    float* w = (float*)d_ws;
    float* ws_fes   = w;                 // 256*128
    float* ws_fet   = w + 32768;
    float* ws_fgs   = w + 65536;
    float* ws_fgt   = w + 98304;
    float* ws_cat_s = w + 131072;        // 256*256
    float* ws_cat_t = w + 196608;
    float* ws_soft_s = w + 262144;       // 256*100 (padded slot)
    float* ws_soft_t = w + 294912;
    float* ws_wn_s   = w + 327680;       // 256
    float* ws_wn_t   = w + 327936;
    int*   ws_nbr_s  = (int*)(w + 328192);  // 256*8
    int*   ws_nbr_t  = (int*)(w + 330240);
    float* ws_pos    = w + 332288;       // 4*256

    zero_kernel<<<1, 32, 0, stream>>>(out);

    // Embeddings (fp32 WMMA GEMM + bias + L2 norm)
    gemm_l2norm_kernel<true><<<16, 256, 0, stream>>>(f_s, W_es, b_es, ws_fes, 1024);
    gemm_l2norm_kernel<true><<<16, 256, 0, stream>>>(f_t, W_et, b_et, ws_fet, 2048);

    // kNN graphs from logits
    softmax_norm_kernel<<<256, 128, 0, stream>>>(l_s, ws_soft_s, ws_wn_s);
    softmax_norm_kernel<<<256, 128, 0, stream>>>(l_t, ws_soft_t, ws_wn_t);
    knn_kernel<<<256, 256, 0, stream>>>(ws_soft_s, ws_wn_s, ws_nbr_s);
    knn_kernel<<<256, 256, 0, stream>>>(ws_soft_t, ws_wn_t, ws_nbr_t);

    // GNN encode: concat([h, mean-neighbor]) @ W + b, L2 norm
    gnn_cat_kernel<<<256, 128, 0, stream>>>(ws_fes, ws_nbr_s, ws_cat_s);
    gnn_cat_kernel<<<256, 128, 0, stream>>>(ws_fet, ws_nbr_t, ws_cat_t);
    gemm_l2norm_kernel<false><<<16, 256, 0, stream>>>(ws_cat_s, W_gs, b_gs, ws_fgs, 256);
    gemm_l2norm_kernel<false><<<16, 256, 0, stream>>>(ws_cat_t, W_gt, b_gt, ws_fgt, 256);

    // Positives (incl. smoothed, detached f_sgs/f_sgt)
    smooth_pos_kernel<<<256, 128, 0, stream>>>(mem_l, mem_ab, idx,
                                               ws_fes, ws_fet, ws_fgs, ws_fgt,
                                               ws_pos);

    // Memory-bound neg gather + streaming logsumexp + loss
    neg_loss_kernel<<<256, 256, 0, stream>>>(mem_l, mem_ab, cidx,
                                             ws_fes, ws_fet, ws_fgs, ws_fgt,
                                             ws_pos, out);
}